// BGA_19550691131810
// MI455X (gfx1250) — compile-verified
//
#include <hip/hip_runtime.h>
#include <hip/hip_bf16.h>
#include <math.h>

typedef __attribute__((ext_vector_type(2))) float v2f;
typedef __attribute__((ext_vector_type(8))) float v8f;

#define EPS_BN 1e-5f

// ---------------------------------------------------------------------------
// Wave-level WMMA GEMM over a 16x16 tile grid.
//   Y[M x Nc] = act( X[M x Kpad] * Wf^T + bf )   (Wf stored [Nc][Kpad] row-major)
// X may live in LDS or global (generic pointer). Kpad % 4 == 0, M % 16 == 0,
// Nc % 16 == 0. If FUSE_MAX: skip Y store, ReLU then column-max into gmax via
// LDS atomicMax on float bits (all values >= 0 after ReLU).
// ---------------------------------------------------------------------------
template <bool RELU, bool FUSE_MAX>
__device__ __forceinline__ void layer_gemm(const float* __restrict__ X, int ldx,
                                           int Mrows,
                                           const float* __restrict__ Wf,
                                           const float* __restrict__ bf,
                                           int Kpad, int Nc,
                                           float* __restrict__ Y,
                                           float* __restrict__ gmax) {
  const int tid    = threadIdx.x;
  const int wave   = tid >> 5;
  const int lane   = tid & 31;
  const int nwaves = blockDim.x >> 5;
  const int half   = lane >> 4;     // 0: lanes 0-15, 1: lanes 16-31
  const int l      = lane & 15;
  const int tilesM = Mrows >> 4;
  const int tilesN = Nc >> 4;
  for (int t = wave; t < tilesM * tilesN; t += nwaves) {
    const int tm = t / tilesN;
    const int tn = t % tilesN;
    v8f acc = {0.f, 0.f, 0.f, 0.f, 0.f, 0.f, 0.f, 0.f};
    // A: lane row m = tm*16 + l, columns k + 2*half, k + 2*half + 1
    const float* Xa = X + (size_t)(tm * 16 + l) * ldx + 2 * half;
    // B: lane col n = tn*16 + l, rows    k + 2*half, k + 2*half + 1
    const float* Wb = Wf + (size_t)(tn * 16 + l) * Kpad + 2 * half;
    for (int k = 0; k < Kpad; k += 4) {
      v2f a; a[0] = Xa[k]; a[1] = Xa[k + 1];
      v2f b; b[0] = Wb[k]; b[1] = Wb[k + 1];
      acc = __builtin_amdgcn_wmma_f32_16x16x4_f32(false, a, false, b,
                                                  (short)0, acc, false, false);
    }
    const int n = tn * 16 + l;
    const float bias = bf[n];
    if (!FUSE_MAX) {
#pragma unroll
      for (int r = 0; r < 8; ++r) {
        const int m = tm * 16 + half * 8 + r;
        float val = acc[r] + bias;
        if (RELU) val = fmaxf(val, 0.f);
        Y[(size_t)m * Nc + n] = val;
      }
    } else {
      float mx = 0.f;
#pragma unroll
      for (int r = 0; r < 8; ++r) {
        float val = fmaxf(acc[r] + bias, 0.f);  // ReLU
        mx = fmaxf(mx, val);
      }
      atomicMax((int*)(gmax + n), __float_as_int(mx));  // mx >= 0
    }
  }
}

// ---------------------------------------------------------------------------
// Transpose xyz (B,3,N) -> (B,N,3)
// ---------------------------------------------------------------------------
__global__ void transpose_xyz(const float* __restrict__ in,
                              float* __restrict__ out, int B, int N) {
  int i = blockIdx.x * blockDim.x + threadIdx.x;
  int total = B * 3 * N;
  if (i >= total) return;
  int b = i / (3 * N);
  int r = i % (3 * N);
  int c = r / N;
  int n = r % N;
  out[((size_t)b * N + n) * 3 + c] = in[i];
}

// ---------------------------------------------------------------------------
// Fold BN into conv/fc weights:  Wf[n][k] = W[n][k] * g[n]/sqrt(v[n]+eps)
//                                bf[n]    = (b[n]-m[n])*s + bt[n]
// Zero-padded to [CoutPad][Kpad].
// ---------------------------------------------------------------------------
__global__ void fold_kernel(const float* __restrict__ W,
                            const float* __restrict__ b,
                            const float* __restrict__ g,
                            const float* __restrict__ bt,
                            const float* __restrict__ m,
                            const float* __restrict__ v,
                            int Cin, int Kpad, int Cout, int CoutPad,
                            int has_bn, float* __restrict__ Wf,
                            float* __restrict__ bf) {
  int i = blockIdx.x * blockDim.x + threadIdx.x;
  int total = CoutPad * Kpad;
  if (i >= total) return;
  int n = i / Kpad;
  int k = i % Kpad;
  float s = 1.f;
  if (n < Cout && has_bn) s = g[n] * rsqrtf(v[n] + EPS_BN);
  Wf[i] = (n < Cout && k < Cin) ? W[n * Cin + k] * s : 0.f;
  if (k == 0) {
    float bb = 0.f;
    if (n < Cout) bb = has_bn ? (b[n] - m[n]) * s + bt[n] : b[n];
    bf[n] = bb;
  }
}

// ---------------------------------------------------------------------------
// Farthest point sampling; one block per batch, distances in LDS.
// ---------------------------------------------------------------------------
__global__ __launch_bounds__(256) void fps_kernel(const float* __restrict__ pts,
                                                  int N, int npoint,
                                                  int* __restrict__ fidx,
                                                  float* __restrict__ newxyz) {
  __shared__ float dist[4096];
  __shared__ float rv[256];
  __shared__ int ri[256];
  __shared__ float cen[3];
  __shared__ int curF;
  const int b = blockIdx.x;
  const int tid = threadIdx.x;
  for (int i = tid; i < N; i += 256) dist[i] = 1e10f;
  if (tid == 0) curF = 0;
  __syncthreads();
  for (int it = 0; it < npoint; ++it) {
    const int f = curF;
    if (tid < 3) {
      float c = pts[((size_t)b * N + f) * 3 + tid];
      cen[tid] = c;
      newxyz[((size_t)b * npoint + it) * 3 + tid] = c;
    }
    if (tid == 0) fidx[b * npoint + it] = f;
    __syncthreads();
    const float cx = cen[0], cy = cen[1], cz = cen[2];
    float bestv = -1.f;
    int besti = 0;
    for (int i = tid; i < N; i += 256) {
      const float* p = pts + ((size_t)b * N + i) * 3;
      float dx = p[0] - cx, dy = p[1] - cy, dz = p[2] - cz;
      float d = dx * dx + dy * dy + dz * dz;
      float nd = fminf(dist[i], d);
      dist[i] = nd;
      if (nd > bestv) { bestv = nd; besti = i; }
    }
    rv[tid] = bestv;
    ri[tid] = besti;
    __syncthreads();
    for (int off = 128; off > 0; off >>= 1) {
      if (tid < off) {
        if (rv[tid + off] > rv[tid] ||
            (rv[tid + off] == rv[tid] && ri[tid + off] < ri[tid])) {
          rv[tid] = rv[tid + off];
          ri[tid] = ri[tid + off];
        }
      }
      __syncthreads();
    }
    if (tid == 0) curF = ri[0];
    __syncthreads();
  }
}

// ---------------------------------------------------------------------------
// Ball query: first 64 indices (ascending) within radius; fill with first hit.
// ---------------------------------------------------------------------------
__global__ void ball_kernel(const float* __restrict__ pts,
                            const float* __restrict__ newxyz, int N, int S,
                            int total, float r2, int* __restrict__ outidx) {
  int g = blockIdx.x * blockDim.x + threadIdx.x;
  if (g >= total) return;
  int b = g / S;
  float cx = newxyz[g * 3 + 0], cy = newxyz[g * 3 + 1], cz = newxyz[g * 3 + 2];
  int* out = outidx + (size_t)g * 64;
  int cnt = 0, first = -1;
  for (int j = 0; j < N && cnt < 64; ++j) {
    const float* p = pts + ((size_t)b * N + j) * 3;
    float dx = p[0] - cx, dy = p[1] - cy, dz = p[2] - cz;
    float d = dx * dx + dy * dy + dz * dz;
    if (d <= r2) {
      if (first < 0) first = j;
      out[cnt++] = j;
    }
  }
  int fill = (first >= 0) ? first : (N - 1);
  for (int k = cnt; k < 64; ++k) out[k] = fill;
}

// ---------------------------------------------------------------------------
// Fused SA stage: gather group -> 3 x (WMMA GEMM + BN-folded bias + ReLU)
// -> max-pool over samples. One block (4 wave32) per output point.
// M chunked (MCHUNK rows x NCH chunks) to bound LDS.
// ---------------------------------------------------------------------------
template <int MCHUNK, int NCH, int K0PAD, int FEATC, int C1, int C2, int C3,
          bool GROUPALL>
__global__ __launch_bounds__(128) void sa_kernel(
    const float* __restrict__ pts, const float* __restrict__ feats,
    const int* __restrict__ ballidx, const float* __restrict__ newxyz,
    const float* __restrict__ Wf1, const float* __restrict__ bf1,
    const float* __restrict__ Wf2, const float* __restrict__ bf2,
    const float* __restrict__ Wf3, const float* __restrict__ bf3,
    float* __restrict__ outfeat, int N, int S) {
  constexpr int EVSZ = (K0PAD > C2) ? MCHUNK * K0PAD : MCHUNK * C2;
  __shared__ float bufE[EVSZ];         // X0 / Y2
  __shared__ float bufO[MCHUNK * C1];  // Y1
  __shared__ float gmax[C3];
  const int tid = threadIdx.x;
  const int bs = blockIdx.x;  // b*S + s
  const int b = bs / S;
  for (int i = tid; i < C3; i += blockDim.x) gmax[i] = 0.f;
  float cx = 0.f, cy = 0.f, cz = 0.f;
  if (!GROUPALL) {
    cx = newxyz[bs * 3 + 0];
    cy = newxyz[bs * 3 + 1];
    cz = newxyz[bs * 3 + 2];
  }
  __syncthreads();
  for (int ch = 0; ch < NCH; ++ch) {
    // gather chunk into bufE (row-major, ld = K0PAD, zero-padded)
    for (int i = tid; i < MCHUNK * K0PAD; i += blockDim.x) {
      int r = i / K0PAD, c = i % K0PAD;
      int j = GROUPALL ? (ch * MCHUNK + r) : ballidx[(size_t)bs * 64 + ch * MCHUNK + r];
      float val = 0.f;
      if (c < 3) {
        val = pts[((size_t)b * N + j) * 3 + c];
        if (!GROUPALL) val -= (c == 0 ? cx : (c == 1 ? cy : cz));
      } else if (c < 3 + FEATC) {
        val = feats[((size_t)b * N + j) * FEATC + (c - 3)];
      }
      bufE[i] = val;
    }
    __syncthreads();
    layer_gemm<true, false>(bufE, K0PAD, MCHUNK, Wf1, bf1, K0PAD, C1, bufO, nullptr);
    __syncthreads();
    layer_gemm<true, false>(bufO, C1, MCHUNK, Wf2, bf2, C1, C2, bufE, nullptr);
    __syncthreads();
    layer_gemm<true, true>(bufE, C2, MCHUNK, Wf3, bf3, C2, C3, nullptr, gmax);
    __syncthreads();
  }
  for (int i = tid; i < C3; i += blockDim.x) outfeat[(size_t)bs * C3 + i] = gmax[i];
}

// ---------------------------------------------------------------------------
// Head: fc1(BN,ReLU) -> fc2(BN,ReLU) -> fc3 -> log_softmax; also emit gfeat.
// One block, 4 wave32, all WMMA.
// ---------------------------------------------------------------------------
__global__ __launch_bounds__(128) void head_kernel(
    const float* __restrict__ gfeat,
    const float* __restrict__ Wf1, const float* __restrict__ bf1,
    const float* __restrict__ Wf2, const float* __restrict__ bf2,
    const float* __restrict__ Wf3, const float* __restrict__ bf3,
    float* __restrict__ out) {
  __shared__ float h1[16 * 512];
  __shared__ float h2[16 * 256];
  __shared__ float lg[16 * 48];
  layer_gemm<true, false>(gfeat, 1024, 16, Wf1, bf1, 1024, 512, h1, nullptr);
  __syncthreads();
  layer_gemm<true, false>(h1, 512, 16, Wf2, bf2, 512, 256, h2, nullptr);
  __syncthreads();
  layer_gemm<false, false>(h2, 256, 16, Wf3, bf3, 256, 48, lg, nullptr);
  __syncthreads();
  const int tid = threadIdx.x;
  if (tid < 16) {
    float mx = -1e30f;
    for (int j = 0; j < 40; ++j) mx = fmaxf(mx, lg[tid * 48 + j]);
    float s = 0.f;
    for (int j = 0; j < 40; ++j) s += expf(lg[tid * 48 + j] - mx);
    float lse = mx + logf(s);
    for (int j = 0; j < 40; ++j) out[tid * 40 + j] = lg[tid * 48 + j] - lse;
  }
  for (int i = tid; i < 16 * 1024; i += blockDim.x) out[640 + i] = gfeat[i];
}

// ---------------------------------------------------------------------------
// Host orchestration
// ---------------------------------------------------------------------------
static inline size_t align256(size_t x) { return (x + 255) & ~(size_t)255; }

extern "C" void kernel_launch(void* const* d_in, const int* in_sizes, int n_in,
                              void* d_out, int out_size, void* d_ws,
                              size_t ws_size, hipStream_t stream) {
  (void)in_sizes; (void)n_in; (void)out_size; (void)ws_size;
  const int B = 16, N = 4096;

  char* base = (char*)d_ws;
  size_t off = 0;
  auto alloc = [&](size_t bytes) -> void* {
    void* r = base + off;
    off = align256(off + bytes);
    return r;
  };

  float* xyzT   = (float*)alloc((size_t)B * N * 3 * 4);
  int*   fidx1  = (int*)  alloc((size_t)B * 512 * 4);
  float* nxyz1  = (float*)alloc((size_t)B * 512 * 3 * 4);
  int*   bidx1  = (int*)  alloc((size_t)B * 512 * 64 * 4);
  float* feat1  = (float*)alloc((size_t)B * 512 * 128 * 4);
  int*   fidx2  = (int*)  alloc((size_t)B * 128 * 4);
  float* nxyz2  = (float*)alloc((size_t)B * 128 * 3 * 4);
  int*   bidx2  = (int*)  alloc((size_t)B * 128 * 64 * 4);
  float* feat2  = (float*)alloc((size_t)B * 128 * 256 * 4);
  float* gfeat  = (float*)alloc((size_t)B * 1024 * 4);

  // Layer table: {Cin, Kpad, Cout, CoutPad, has_bn, input-base index}
  // d_in layout: 0 xyz; sa1 layers (W,b,g,bt,m,v) at 1,7,13; sa2 at 19,25,31;
  // sa3 at 37,43,49; head: fc1W 55, fc1b 56, bn1 57-60, fc2W 61, fc2b 62,
  // bn2 63-66, fc3W 67, fc3b 68.
  struct L { int cin, kpad, cout, coutpad, bn, iw, ib, ig, ibt, im, iv; };
  const L layers[12] = {
      {3, 4, 64, 64, 1, 1, 2, 3, 4, 5, 6},
      {64, 64, 64, 64, 1, 7, 8, 9, 10, 11, 12},
      {64, 64, 128, 128, 1, 13, 14, 15, 16, 17, 18},
      {131, 132, 128, 128, 1, 19, 20, 21, 22, 23, 24},
      {128, 128, 128, 128, 1, 25, 26, 27, 28, 29, 30},
      {128, 128, 256, 256, 1, 31, 32, 33, 34, 35, 36},
      {259, 260, 256, 256, 1, 37, 38, 39, 40, 41, 42},
      {256, 256, 512, 512, 1, 43, 44, 45, 46, 47, 48},
      {512, 512, 1024, 1024, 1, 49, 50, 51, 52, 53, 54},
      {1024, 1024, 512, 512, 1, 55, 56, 57, 58, 59, 60},
      {512, 512, 256, 256, 1, 61, 62, 63, 64, 65, 66},
      {256, 256, 40, 48, 0, 67, 68, -1, -1, -1, -1},
  };
  float* Wf[12];
  float* bf[12];
  for (int i = 0; i < 12; ++i) {
    Wf[i] = (float*)alloc((size_t)layers[i].coutpad * layers[i].kpad * 4);
    bf[i] = (float*)alloc((size_t)layers[i].coutpad * 4);
  }

  // 1) transpose xyz -> (B,N,3)
  {
    int total = B * 3 * N;
    transpose_xyz<<<(total + 255) / 256, 256, 0, stream>>>(
        (const float*)d_in[0], xyzT, B, N);
  }
  // 2) fold BN into all 12 layers
  for (int i = 0; i < 12; ++i) {
    const L& L_ = layers[i];
    int total = L_.coutpad * L_.kpad;
    fold_kernel<<<(total + 255) / 256, 256, 0, stream>>>(
        (const float*)d_in[L_.iw], (const float*)d_in[L_.ib],
        L_.bn ? (const float*)d_in[L_.ig] : nullptr,
        L_.bn ? (const float*)d_in[L_.ibt] : nullptr,
        L_.bn ? (const float*)d_in[L_.im] : nullptr,
        L_.bn ? (const float*)d_in[L_.iv] : nullptr,
        L_.cin, L_.kpad, L_.cout, L_.coutpad, L_.bn, Wf[i], bf[i]);
  }
  // 3) SA1: FPS 4096->512, ball r=0.2, MLP 4->64->64->128, max64
  fps_kernel<<<B, 256, 0, stream>>>(xyzT, N, 512, fidx1, nxyz1);
  ball_kernel<<<(B * 512 + 127) / 128, 128, 0, stream>>>(
      xyzT, nxyz1, N, 512, B * 512, 0.2f * 0.2f, bidx1);
  sa_kernel<64, 1, 4, 0, 64, 64, 128, false><<<B * 512, 128, 0, stream>>>(
      xyzT, nullptr, bidx1, nxyz1, Wf[0], bf[0], Wf[1], bf[1], Wf[2], bf[2],
      feat1, N, 512);
  // 4) SA2: FPS 512->128 on nxyz1, ball r=0.4, MLP 132->128->128->256, max64
  fps_kernel<<<B, 256, 0, stream>>>(nxyz1, 512, 128, fidx2, nxyz2);
  ball_kernel<<<(B * 128 + 127) / 128, 128, 0, stream>>>(
      nxyz1, nxyz2, 512, 128, B * 128, 0.4f * 0.4f, bidx2);
  sa_kernel<32, 2, 132, 128, 128, 128, 256, false><<<B * 128, 128, 0, stream>>>(
      nxyz1, feat1, bidx2, nxyz2, Wf[3], bf[3], Wf[4], bf[4], Wf[5], bf[5],
      feat2, 512, 128);
  // 5) SA3: group_all over 128 points, MLP 260->256->512->1024, max128
  sa_kernel<16, 8, 260, 256, 256, 512, 1024, true><<<B, 128, 0, stream>>>(
      nxyz2, feat2, nullptr, nullptr, Wf[6], bf[6], Wf[7], bf[7], Wf[8], bf[8],
      gfeat, 128, 1);
  // 6) Head + log_softmax + gfeat copy
  head_kernel<<<1, 128, 0, stream>>>(gfeat, Wf[9], bf[9], Wf[10], bf[10],
                                     Wf[11], bf[11], (float*)d_out);
}